// PagedAttention_33990371180575
// MI455X (gfx1250) — compile-verified
//
#include <hip/hip_runtime.h>
#include <math.h>

// Problem constants (from reference)
#define Bn 16
#define Hn 32
#define KVHn 8
#define Gn 4            // H / KVH
#define Dn 128
#define Sn 4096
#define SPLITS 8
#define SCHUNK (Sn / SPLITS)   // 512 slots per split
#define TS 16                  // slot tile (WMMA M)
#define NW 4                   // waves per block
#define NTHREADS (NW * 32)
#define LOG_BASE 9.210340371976184f     // ln(10000)
#define SCALEF 0.08838834764831845f     // 1/sqrt(128)

typedef float v2f __attribute__((ext_vector_type(2)));
typedef float v4f __attribute__((ext_vector_type(4)));
typedef float v8f __attribute__((ext_vector_type(8)));

// ---------------------------------------------------------------------------
// Kernel 1: per-(b, kvh, split) partial attention (flash-decoding style).
// 4 waves / block. WMMA f32 16x16x4 for both QK^T and P*V.
// ---------------------------------------------------------------------------
__global__ __launch_bounds__(NTHREADS) void attn_partial(
    const float* __restrict__ q, const float* __restrict__ kc,
    const float* __restrict__ vc, const int* __restrict__ slots,
    const int* __restrict__ pos, const int* __restrict__ clen,
    float* __restrict__ msc, float* __restrict__ lsc, float* __restrict__ osc)
{
  __shared__ float sINV[64];                 // RoPE inverse frequencies
  __shared__ float sQ[Gn][Dn];               // RoPE'd query (4 x 128)
  __shared__ float sP[TS][SCHUNK];           // scores->probs, rows G..15 stay 0
  __shared__ float sKV[NW][TS][Dn + 4];      // per-wave K/V staging, padded rows
  __shared__ float sO[Gn][Dn];               // cross-wave output accumulation

  const int blk   = blockIdx.x;
  const int split = blk % SPLITS;
  const int bk    = blk / SPLITS;
  const int kvh   = bk % KVHn;
  const int b     = bk / KVHn;

  const int tid  = threadIdx.x;
  const int w    = tid >> 5;      // wave id 0..3
  const int lane = tid & 31;
  const int half = lane >> 4;     // WMMA lane half
  const int nidx = lane & 15;     // WMMA N / M-row index
  const int lrow = lane >> 1;     // staging: 2 lanes per matrix row
  const int lh   = lane & 1;      // staging: which half of D

  // ---- Phase 0: inv_freq, zero accum + pad rows, RoPE the 4 query heads ----
  if (tid < 64) sINV[tid] = __expf((float)tid * (-2.0f / (float)Dn) * LOG_BASE);
  for (int i = tid; i < Gn * Dn; i += NTHREADS) (&sO[0][0])[i] = 0.0f;
  {
    float* ppad = &sP[Gn][0];   // rows Gn..TS-1, contiguous
    for (int i = tid; i < (TS - Gn) * SCHUNK; i += NTHREADS) ppad[i] = 0.0f;
  }
  __syncthreads();

  const int ctx   = clen[b];
  const int plast = pos[(size_t)b * Sn + (ctx - 1)];
  for (int pp = tid; pp < Gn * 64; pp += NTHREADS) {
    const int g = pp >> 6, j = pp & 63;
    const float ang = (float)plast * sINV[j];
    const float c = __cosf(ang), s = __sinf(ang);
    const float* qrow = q + ((size_t)b * Hn + kvh * Gn + g) * Dn;
    const float x1 = qrow[j], x2 = qrow[j + 64];
    sQ[g][j]      = x1 * c - x2 * s;
    sQ[g][j + 64] = x2 * c + x1 * s;
  }
  __syncthreads();

  // Hoist loop-invariant q B-fragments into registers (one predicated pass).
  // B layout: vgpr0 lanes0-15 K=k0, lanes16-31 K=k0+2; vgpr1 K=k0+1 / k0+3.
  v2f qb[Dn / 4];
#pragma unroll
  for (int k0 = 0; k0 < Dn; k0 += 4) {
    const int kk = k0 + 2 * half;
    v2f t = {0.0f, 0.0f};
    if (nidx < Gn) t = *(const v2f*)&sQ[nidx][kk];
    qb[k0 >> 2] = t;
  }

  // ---- Phase A: scores.  D(16 slots x 16 cols) = Kr(16x128) * qr^T(128x16) ----
  for (int t = w; t < SCHUNK / TS; t += NW) {
    const int sbase = split * SCHUNK + t * TS;
    {
      const int srow = sbase + lrow;
      const int slot = slots[(size_t)b * Sn + srow];
      const int p    = pos[(size_t)b * Sn + srow];
      const float* krow = kc + ((size_t)slot * KVHn + kvh) * Dn;
      // warm L2 for the V row this block reads in Phase C
      __builtin_prefetch(vc + ((size_t)slot * KVHn + kvh) * Dn + lh * 64, 0, 1);
      float* dst = &sKV[w][lrow][0];
#pragma unroll
      for (int jj = 0; jj < 8; ++jj) {
        const int d0 = lh * 32 + jj * 4;
        v4f x1 = *(const v4f*)(krow + d0);
        v4f x2 = *(const v4f*)(krow + 64 + d0);
        v4f c, s;
#pragma unroll
        for (int e = 0; e < 4; ++e) {
          const float ang = (float)p * sINV[d0 + e];
          c[e] = __cosf(ang);
          s[e] = __sinf(ang);
        }
        *(v4f*)(dst + d0)      = x1 * c - x2 * s;
        *(v4f*)(dst + 64 + d0) = x2 * c + x1 * s;
      }
    }
    asm volatile("s_wait_dscnt 0" ::: "memory");   // wave-local staging ready

    v8f acc0 = {};
    v8f acc1 = {};
#pragma unroll
    for (int k0 = 0; k0 < Dn; k0 += 8) {
      const int kk = k0 + 2 * half;
      // A fragment: Kr rows (M = slot), ISA 32-bit A layout
      v2f a0 = *(const v2f*)&sKV[w][nidx][kk];
      v2f a1 = *(const v2f*)&sKV[w][nidx][kk + 4];
      acc0 = __builtin_amdgcn_wmma_f32_16x16x4_f32(false, a0, false, qb[k0 >> 2],
                                                   (short)0, acc0, false, false);
      acc1 = __builtin_amdgcn_wmma_f32_16x16x4_f32(false, a1, false, qb[(k0 >> 2) + 1],
                                                   (short)0, acc1, false, false);
    }
    // C layout: VGPR r, lanes0-15 -> M=r, lanes16-31 -> M=r+8, N=lane&15
    if (nidx < Gn) {
#pragma unroll
      for (int rr = 0; rr < 8; ++rr) {
        const int m = rr + 8 * half;
        float v = (acc0[rr] + acc1[rr]) * SCALEF;
        v = (sbase + m < ctx) ? v : -1e30f;
        sP[nidx][t * TS + m] = v;
      }
    }
  }
  __syncthreads();

  // ---- Phase B: per-head softmax stats over this split (wave w <-> head w) ----
  {
    const int g = w;
    float m = -INFINITY;
    for (int i = lane; i < SCHUNK; i += 32) m = fmaxf(m, sP[g][i]);
#pragma unroll
    for (int off = 16; off > 0; off >>= 1) m = fmaxf(m, __shfl_xor(m, off));
    float l = 0.0f;
    for (int i = lane; i < SCHUNK; i += 32) {
      const float pz = __expf(sP[g][i] - m);
      sP[g][i] = pz;
      l += pz;
    }
#pragma unroll
    for (int off = 16; off > 0; off >>= 1) l += __shfl_xor(l, off);
    if (lane == 0) {
      const size_t idx = (((size_t)b * KVHn + kvh) * SPLITS + split) * Gn + g;
      msc[idx] = m;
      lsc[idx] = l;
    }
  }
  __syncthreads();

  // ---- Phase C: out = P^T * V via WMMA, 8 N-tiles covering D=128 ----
  v8f oacc[8] = {};
  for (int t = w; t < SCHUNK / TS; t += NW) {
    const int pb = t * TS;
    {
      const int srow = split * SCHUNK + pb + lrow;
      const int slot = slots[(size_t)b * Sn + srow];
      const float* vrow = vc + ((size_t)slot * KVHn + kvh) * Dn;
      float* dst = &sKV[w][lrow][0];
#pragma unroll
      for (int jj = 0; jj < 16; ++jj) {
        const int d0 = lh * 64 + jj * 4;
        *(v4f*)(dst + d0) = *(const v4f*)(vrow + d0);
      }
    }
    asm volatile("s_wait_dscnt 0" ::: "memory");
#pragma unroll
    for (int k0 = 0; k0 < TS; k0 += 4) {
      const int kk = k0 + 2 * half;
      // A fragment: P^T (M = head); rows >= G are exact zeros (padded sP)
      v2f av = *(const v2f*)&sP[nidx][pb + kk];
#pragma unroll
      for (int n0 = 0; n0 < 8; ++n0) {
        v2f bv;
        bv.x = sKV[w][kk][n0 * 16 + nidx];
        bv.y = sKV[w][kk + 1][n0 * 16 + nidx];
        oacc[n0] = __builtin_amdgcn_wmma_f32_16x16x4_f32(false, av, false, bv,
                                                         (short)0, oacc[n0],
                                                         false, false);
      }
    }
  }
  // cross-wave reduction: valid heads live in vgpr rr<G, lanes 0-15
  if (half == 0) {
#pragma unroll
    for (int n0 = 0; n0 < 8; ++n0)
#pragma unroll
      for (int rr = 0; rr < Gn; ++rr)
        atomicAdd(&sO[rr][n0 * 16 + nidx], oacc[n0][rr]);
  }
  __syncthreads();

  float* obase = osc + ((((size_t)b * KVHn + kvh) * SPLITS + split) * Gn) * Dn;
  for (int i = tid; i < Gn * Dn; i += NTHREADS) obase[i] = (&sO[0][0])[i];
}

// ---------------------------------------------------------------------------
// Kernel 2: merge the SPLITS partial results (rescale by exp(m_i - m*), /l*)
// ---------------------------------------------------------------------------
__global__ __launch_bounds__(Dn) void attn_merge(
    const float* __restrict__ msc, const float* __restrict__ lsc,
    const float* __restrict__ osc, float* __restrict__ out)
{
  const int bh = blockIdx.x;       // over B*H
  const int b = bh / Hn, h = bh % Hn;
  const int kvh = h / Gn, g = h % Gn;
  const int d = threadIdx.x;       // 0..127

  const size_t base = ((size_t)b * KVHn + kvh) * SPLITS;
  float mmax = -INFINITY;
#pragma unroll
  for (int i = 0; i < SPLITS; ++i)
    mmax = fmaxf(mmax, msc[(base + i) * Gn + g]);
  float num = 0.0f, den = 0.0f;
#pragma unroll
  for (int i = 0; i < SPLITS; ++i) {
    const float wgt = __expf(msc[(base + i) * Gn + g] - mmax);
    den += lsc[(base + i) * Gn + g] * wgt;
    num += osc[((base + i) * Gn + g) * Dn + d] * wgt;
  }
  out[((size_t)b * Hn + h) * Dn + d] = num / den;
}

// ---------------------------------------------------------------------------
extern "C" void kernel_launch(void* const* d_in, const int* in_sizes, int n_in,
                              void* d_out, int out_size, void* d_ws, size_t ws_size,
                              hipStream_t stream) {
  (void)in_sizes; (void)n_in; (void)out_size; (void)ws_size;
  const float* q     = (const float*)d_in[0];
  const float* kc    = (const float*)d_in[1];
  const float* vc    = (const float*)d_in[2];
  const int*   slots = (const int*)d_in[3];
  const int*   pos   = (const int*)d_in[4];
  const int*   clen  = (const int*)d_in[5];
  float* out = (float*)d_out;

  // scratch: m (4K floats) | l (4K floats) | partial out (512K floats ~ 2MB)
  float* msc = (float*)d_ws;
  float* lsc = msc + (size_t)Bn * KVHn * SPLITS * Gn;
  float* osc = lsc + (size_t)Bn * KVHn * SPLITS * Gn;

  attn_partial<<<dim3(Bn * KVHn * SPLITS), dim3(NTHREADS), 0, stream>>>(
      q, kc, vc, slots, pos, clen, msc, lsc, osc);
  attn_merge<<<dim3(Bn * Hn), dim3(Dn), 0, stream>>>(msc, lsc, osc, out);
}